// Pointnet2MSG_46789373723401
// MI455X (gfx1250) — compile-verified
//
#include <hip/hip_runtime.h>
#include <hip/hip_bf16.h>
#include <math.h>

#define BB 4
#define NPTS 8192
#define TOPKK 512

typedef __attribute__((ext_vector_type(2))) float v2f;
typedef __attribute__((ext_vector_type(8))) float v8f;

static inline int cdiv(int a, int b) { return (a + b - 1) / b; }

// ---------------------------------------------------------------------------
// Generic WMMA fp32 GEMM:  C[M,Nc] = act(A[M,K] * W[K,Nc] + bias)
// act: 0 = none, 1 = relu, 2 = leaky(0.1)
// One wave computes one 16x16 output tile using V_WMMA_F32_16X16X4_F32.
// A-layout (16x4 f32): lanes 0-15 hold K={k0,k0+1}, lanes 16-31 hold K={k0+2,k0+3}
// B-layout (4x16 f32): lane%16 = column, lane-half selects K pair (symmetric)
// C/D: 8 VGPRs, M = v + 8*(lane>=16), N = lane&15
//
// Range handling: out-of-range A rows / B cols are CLAMPED (their garbage only
// reaches C rows/cols that are never stored), so the main K loop is fully
// branch-free. Only the K remainder needs zero-fill (single guarded tail).
// ---------------------------------------------------------------------------
__global__ void wmma_gemm_bias_act(const float* __restrict__ A,
                                   const float* __restrict__ W,
                                   const float* __restrict__ bias,
                                   float* __restrict__ C,
                                   int M, int K, int Nc, int act)
{
    const int lane = threadIdx.x;      // 0..31
    const int wave = threadIdx.y;      // 0..7
    const int tilesN = (Nc + 15) >> 4;
    const int tilesM = (M + 15) >> 4;
    const int tile = blockIdx.x * 8 + wave;
    if (tile >= tilesM * tilesN) return;
    const int tm = tile / tilesN;
    const int tn = tile % tilesN;

    const int half = lane >> 4;        // 0: lanes 0-15, 1: lanes 16-31
    const int l16  = lane & 15;
    const int arow = tm * 16 + l16;    // A row owned by this lane
    const int bcol = tn * 16 + l16;    // W column owned by this lane
    const int arow_c = (arow < M) ? arow : (M - 1);   // clamped: always in-bounds
    const int bcol_c = (bcol < Nc) ? bcol : (Nc - 1);

    const float* __restrict__ ap = A + (size_t)arow_c * K + half * 2; // lane's K base
    const float* __restrict__ wp = W + bcol_c;                        // lane's column
    const size_t wstep = (size_t)Nc;

    v8f acc = {0.f, 0.f, 0.f, 0.f, 0.f, 0.f, 0.f, 0.f};

    const int Kmain = K & ~3;
    int k0 = 0;
    #pragma unroll 4
    for (; k0 < Kmain; k0 += 4) {
        v2f a, b;
        a.x = ap[k0 + 0];
        a.y = ap[k0 + 1];
        b.x = wp[(size_t)(k0 + half * 2 + 0) * wstep];
        b.y = wp[(size_t)(k0 + half * 2 + 1) * wstep];
        acc = __builtin_amdgcn_wmma_f32_16x16x4_f32(
            false, a, false, b, (short)0, acc, false, false);
    }
    if (k0 < K) {                       // single guarded tail (K % 4 != 0)
        const int ka = k0 + half * 2;
        v2f a, b;
        a.x = (ka + 0 < K) ? ap[k0 + 0] : 0.f;
        a.y = (ka + 1 < K) ? ap[k0 + 1] : 0.f;
        b.x = (ka + 0 < K) ? wp[(size_t)(ka + 0) * wstep] : 0.f;
        b.y = (ka + 1 < K) ? wp[(size_t)(ka + 1) * wstep] : 0.f;
        acc = __builtin_amdgcn_wmma_f32_16x16x4_f32(
            false, a, false, b, (short)0, acc, false, false);
    }

    if (bcol < Nc) {
        const float bv = bias ? bias[bcol] : 0.f;
        #pragma unroll
        for (int v = 0; v < 8; ++v) {
            const int row = tm * 16 + v + half * 8;
            if (row < M) {
                float x = acc[v] + bv;
                if (act == 1)      x = fmaxf(x, 0.f);
                else if (act == 2) x = (x > 0.f) ? x : 0.1f * x;
                C[(size_t)row * Nc + bcol] = x;
            }
        }
    }
}

// ---------------------------------------------------------------------------
// Farthest point sampling. One block per batch; shared distance array.
// Matches reference: first index 0, argmax with first-max (lowest index) tie.
// ---------------------------------------------------------------------------
__global__ void fps_kernel(const float* __restrict__ xyz, int* __restrict__ fidx,
                           int n, int npoint)
{
    __shared__ float sdist[NPTS];
    __shared__ float rv[256];
    __shared__ int   ri[256];
    __shared__ int   s_last;
    const int b = blockIdx.x, t = threadIdx.x;
    const float* px = xyz + (size_t)b * n * 3;
    for (int j = t; j < n; j += 256) sdist[j] = 1e10f;
    if (t == 0) { fidx[(size_t)b * npoint] = 0; s_last = 0; }
    __syncthreads();
    for (int i = 1; i < npoint; ++i) {
        const int last = s_last;
        const float lx = px[last * 3 + 0];
        const float ly = px[last * 3 + 1];
        const float lz = px[last * 3 + 2];
        float best = -1.f; int bi = 0x7fffffff;
        for (int j = t; j < n; j += 256) {
            const float dx = px[j * 3 + 0] - lx;
            const float dy = px[j * 3 + 1] - ly;
            const float dz = px[j * 3 + 2] - lz;
            const float d = fminf(sdist[j], dx * dx + dy * dy + dz * dz);
            sdist[j] = d;
            if (d > best) { best = d; bi = j; }   // strided ascending -> first max
        }
        rv[t] = best; ri[t] = bi;
        __syncthreads();
        for (int s = 128; s > 0; s >>= 1) {
            if (t < s) {
                if (rv[t + s] > rv[t] || (rv[t + s] == rv[t] && ri[t + s] < ri[t])) {
                    rv[t] = rv[t + s]; ri[t] = ri[t + s];
                }
            }
            __syncthreads();
        }
        if (t == 0) { s_last = ri[0]; fidx[(size_t)b * npoint + i] = ri[0]; }
        __syncthreads();
    }
}

__global__ void gather_xyz_kernel(const float* __restrict__ xyz,
                                  const int* __restrict__ fidx,
                                  float* __restrict__ out, int n, int m)
{
    const int tid = blockIdx.x * blockDim.x + threadIdx.x;
    if (tid >= BB * m) return;
    const int b = tid / m;
    const int j = fidx[tid];
    const float* p = xyz + ((size_t)b * n + j) * 3;
    float* o = out + (size_t)tid * 3;
    o[0] = p[0]; o[1] = p[1]; o[2] = p[2];
}

// ---------------------------------------------------------------------------
// Ball query: first `ns` point indices (ascending) within radius; pad with
// first valid index (or 0 if none). Matches top_k(-key) semantics exactly.
// ---------------------------------------------------------------------------
__global__ void ball_query_kernel(const float* __restrict__ xyz,
                                  const float* __restrict__ new_xyz,
                                  int* __restrict__ idx,
                                  int n, int m, float r2, int ns)
{
    const int tid = blockIdx.x * blockDim.x + threadIdx.x;
    if (tid >= BB * m) return;
    const int b = tid / m;
    const float* px = xyz + (size_t)b * n * 3;
    const float qx = new_xyz[(size_t)tid * 3 + 0];
    const float qy = new_xyz[(size_t)tid * 3 + 1];
    const float qz = new_xyz[(size_t)tid * 3 + 2];
    int* out = idx + (size_t)tid * ns;
    int cnt = 0;
    for (int j = 0; j < n && cnt < ns; ++j) {
        const float dx = px[j * 3 + 0] - qx;
        const float dy = px[j * 3 + 1] - qy;
        const float dz = px[j * 3 + 2] - qz;
        if (dx * dx + dy * dy + dz * dz < r2) out[cnt++] = j;
    }
    const int first = (cnt > 0) ? out[0] : 0;
    for (int k = cnt; k < ns; ++k) out[k] = first;
}

// g[b,q,s,:] = concat(xyz[idx]-center, feats[idx])
__global__ void group_kernel(const float* __restrict__ xyz,
                             const float* __restrict__ feats,
                             const float* __restrict__ new_xyz,
                             const int* __restrict__ idx,
                             float* __restrict__ g,
                             int n, int m, int ns, int cf)
{
    const int tid = blockIdx.x * blockDim.x + threadIdx.x;
    const int total = BB * m * ns;
    if (tid >= total) return;
    const int b = tid / (m * ns);
    const int rem = tid % (m * ns);
    const int q = rem / ns;
    const int j = idx[tid];
    const float* p = xyz + ((size_t)b * n + j) * 3;
    const float* c = new_xyz + ((size_t)b * m + q) * 3;
    float* o = g + (size_t)tid * (3 + cf);
    o[0] = p[0] - c[0]; o[1] = p[1] - c[1]; o[2] = p[2] - c[2];
    if (cf > 0) {
        const float* f = feats + ((size_t)b * n + j) * cf;
        for (int ch = 0; ch < cf; ++ch) o[3 + ch] = f[ch];
    }
}

// out[p, coff+ch] = max_s h[p*ns+s, ch]
__global__ void maxpool_kernel(const float* __restrict__ h, float* __restrict__ out,
                               int np, int ns, int c, int ctot, int coff)
{
    const int tid = blockIdx.x * blockDim.x + threadIdx.x;
    if (tid >= np * c) return;
    const int p = tid / c, ch = tid % c;
    const float* src = h + ((size_t)p * ns) * c + ch;
    float v = src[0];
    for (int s = 1; s < ns; ++s) v = fmaxf(v, src[(size_t)s * c]);
    out[(size_t)p * ctot + coff + ch] = v;
}

// three nearest known neighbors + normalized inverse-distance weights
__global__ void three_nn_kernel(const float* __restrict__ unk,
                                const float* __restrict__ kn,
                                int nu, int nk,
                                float* __restrict__ w3, int* __restrict__ i3)
{
    const int tid = blockIdx.x * blockDim.x + threadIdx.x;
    if (tid >= BB * nu) return;
    const int b = tid / nu;
    const float ux = unk[(size_t)tid * 3 + 0];
    const float uy = unk[(size_t)tid * 3 + 1];
    const float uz = unk[(size_t)tid * 3 + 2];
    const float* pk = kn + (size_t)b * nk * 3;
    float d0 = 1e30f, d1 = 1e30f, d2 = 1e30f;
    int i0 = 0, i1 = 0, i2 = 0;
    for (int j = 0; j < nk; ++j) {
        const float dx = pk[j * 3 + 0] - ux;
        const float dy = pk[j * 3 + 1] - uy;
        const float dz = pk[j * 3 + 2] - uz;
        const float d = dx * dx + dy * dy + dz * dz;
        if (d < d0)      { d2 = d1; i2 = i1; d1 = d0; i1 = i0; d0 = d; i0 = j; }
        else if (d < d1) { d2 = d1; i2 = i1; d1 = d; i1 = j; }
        else if (d < d2) { d2 = d; i2 = j; }
    }
    float w0 = 1.f / (d0 + 1e-8f), w1 = 1.f / (d1 + 1e-8f), w2 = 1.f / (d2 + 1e-8f);
    const float s = w0 + w1 + w2;
    w3[(size_t)tid * 3 + 0] = w0 / s;
    w3[(size_t)tid * 3 + 1] = w1 / s;
    w3[(size_t)tid * 3 + 2] = w2 / s;
    i3[(size_t)tid * 3 + 0] = i0;
    i3[(size_t)tid * 3 + 1] = i1;
    i3[(size_t)tid * 3 + 2] = i2;
}

// h[b,u,:] = concat(sum_j w_j * kn_feats[i_j], unk_feats[b,u])
__global__ void interp_concat_kernel(const float* __restrict__ kn_feats,
                                     const float* __restrict__ w3,
                                     const int* __restrict__ i3,
                                     const float* __restrict__ unk_feats,
                                     float* __restrict__ h,
                                     int nu, int nk, int ck, int cu)
{
    const int tid = blockIdx.x * blockDim.x + threadIdx.x;
    const int ct = ck + cu;
    if (tid >= BB * nu * ct) return;
    const int row = tid / ct;
    const int ch = tid % ct;
    const int b = row / nu;
    float val;
    if (ch < ck) {
        const float* wp = w3 + (size_t)row * 3;
        const int* ip = i3 + (size_t)row * 3;
        const float* kf = kn_feats + (size_t)b * nk * ck;
        val = wp[0] * kf[(size_t)ip[0] * ck + ch]
            + wp[1] * kf[(size_t)ip[1] * ck + ch]
            + wp[2] * kf[(size_t)ip[2] * ck + ch];
    } else {
        val = unk_feats[(size_t)row * cu + (ch - ck)];
    }
    h[(size_t)row * ct + ch] = val;
}

// importance = clip(sigmoid(f @ w + b), 0, 1)
__global__ void att_kernel(const float* __restrict__ f, const float* __restrict__ w,
                           const float* __restrict__ bias, float* __restrict__ imp,
                           int total, int c)
{
    const int tid = blockIdx.x * blockDim.x + threadIdx.x;
    if (tid >= total) return;
    const float* row = f + (size_t)tid * c;
    float s = bias[0];
    for (int i = 0; i < c; ++i) s += row[i] * w[i];
    const float sg = 1.f / (1.f + expf(-s));
    imp[tid] = fminf(fmaxf(sg, 0.f), 1.f);
}

// per-batch top-k by iterative argmax (tie -> lowest index) = lax.top_k order
__global__ void topk_kernel(const float* __restrict__ imp, int* __restrict__ inds,
                            int n, int k)
{
    __shared__ float sv[NPTS];
    __shared__ float rv[256];
    __shared__ int   ri[256];
    const int b = blockIdx.x, t = threadIdx.x;
    for (int j = t; j < n; j += 256) sv[j] = imp[(size_t)b * n + j];
    __syncthreads();
    for (int i = 0; i < k; ++i) {
        float best = -1e30f; int bi = 0x7fffffff;
        for (int j = t; j < n; j += 256) {
            if (sv[j] > best) { best = sv[j]; bi = j; }
        }
        rv[t] = best; ri[t] = bi;
        __syncthreads();
        for (int s = 128; s > 0; s >>= 1) {
            if (t < s) {
                if (rv[t + s] > rv[t] || (rv[t + s] == rv[t] && ri[t + s] < ri[t])) {
                    rv[t] = rv[t + s]; ri[t] = ri[t + s];
                }
            }
            __syncthreads();
        }
        if (t == 0) { inds[(size_t)b * k + i] = ri[0]; sv[ri[0]] = -1e30f; }
        __syncthreads();
    }
}

__global__ void gather_rows_kernel(const float* __restrict__ src,
                                   const int* __restrict__ inds,
                                   float* __restrict__ dst, int n, int k, int c)
{
    const int tid = blockIdx.x * blockDim.x + threadIdx.x;
    if (tid >= BB * k * c) return;
    const int row = tid / c, ch = tid % c;
    const int b = row / k;
    const int j = inds[row];
    dst[tid] = src[((size_t)b * n + j) * c + ch];
}

__global__ void i2f_kernel(const int* __restrict__ src, float* __restrict__ dst, int n)
{
    const int tid = blockIdx.x * blockDim.x + threadIdx.x;
    if (tid < n) dst[tid] = (float)src[tid];
}

__global__ void copyf_kernel(const float* __restrict__ src, float* __restrict__ dst, int n)
{
    const int tid = blockIdx.x * blockDim.x + threadIdx.x;
    if (tid < n) dst[tid] = src[tid];
}

// ---------------------------------------------------------------------------
// Host-side orchestration
// ---------------------------------------------------------------------------
struct Layer { const float* w; const float* b; int cin; int cout; };

static void launch_gemm(const float* A, const Layer& L, float* out, int M, int act,
                        hipStream_t s)
{
    const int tiles = cdiv(M, 16) * cdiv(L.cout, 16);
    dim3 blk(32, 8);
    dim3 grd(cdiv(tiles, 8));
    wmma_gemm_bias_act<<<grd, blk, 0, s>>>(A, L.w, L.b, out, M, L.cin, L.cout, act);
}

// layer outputs ping-pong bufB, bufA, bufB, ...; last goes to dst if non-null
static float* run_chain(const float* in, float* bufA, float* bufB, float* dst,
                        int M, const Layer* L, int nl, int act, hipStream_t s)
{
    const float* cur = in;
    float* out = nullptr;
    for (int i = 0; i < nl; ++i) {
        out = (dst && i == nl - 1) ? dst : ((i % 2 == 0) ? bufB : bufA);
        launch_gemm(cur, L[i], out, M, act, s);
        cur = out;
    }
    return out;
}

static void sa_scale(const float* xyz, const float* feats, int cf,
                     const float* new_xyz, int n, int m, float radius, int ns,
                     const Layer* L, int nl, int* ballidx, float* bufA, float* bufB,
                     float* feat_out, int ctot, int coff, hipStream_t s)
{
    const int q = BB * m;
    ball_query_kernel<<<cdiv(q, 256), 256, 0, s>>>(xyz, new_xyz, ballidx, n, m,
                                                   radius * radius, ns);
    const int rows = BB * m * ns;
    group_kernel<<<cdiv(rows, 256), 256, 0, s>>>(xyz, feats, new_xyz, ballidx, bufA,
                                                 n, m, ns, cf);
    float* h = run_chain(bufA, bufA, bufB, nullptr, rows, L, nl, /*relu*/1, s);
    const int c = L[nl - 1].cout;
    maxpool_kernel<<<cdiv(BB * m * c, 256), 256, 0, s>>>(h, feat_out, BB * m, ns, c,
                                                         ctot, coff);
}

extern "C" void kernel_launch(void* const* d_in, const int* in_sizes, int n_in,
                              void* d_out, int out_size, void* d_ws, size_t ws_size,
                              hipStream_t stream)
{
    (void)in_sizes; (void)n_in; (void)out_size; (void)ws_size;
    const float* pc = (const float*)d_in[0];
    auto F = [&](int i) { return (const float*)d_in[i]; };

    // ---- weights (dict insertion order: sa, fp, imp_hidden, imp_att, fc) ----
    Layer sa0s0[3] = { {F(1),F(2),3,16},   {F(3),F(4),16,16},   {F(5),F(6),16,32} };
    Layer sa0s1[3] = { {F(7),F(8),3,32},   {F(9),F(10),32,32},  {F(11),F(12),32,64} };
    Layer sa1s0[3] = { {F(13),F(14),99,64},{F(15),F(16),64,64}, {F(17),F(18),64,128} };
    Layer sa1s1[3] = { {F(19),F(20),99,64},{F(21),F(22),64,96}, {F(23),F(24),96,128} };
    Layer sa2s0[3] = { {F(25),F(26),259,128},{F(27),F(28),128,256},{F(29),F(30),256,512} };
    Layer sa2s1[3] = { {F(31),F(32),259,128},{F(33),F(34),128,256},{F(35),F(36),256,512} };
    Layer fp0[2] = { {F(37),F(38),1280,256}, {F(39),F(40),256,256} };
    Layer fp1[2] = { {F(41),F(42),352,256},  {F(43),F(44),256,128} };
    Layer fp2[2] = { {F(45),F(46),128,128},  {F(47),F(48),128,128} };
    Layer imp[2] = { {F(49),F(50),128,64},   {F(51),F(52),64,64} };
    const float* att_w = F(53);  // 64x1
    const float* att_b = F(54);  // 1
    Layer fc = { F(55), F(56), 128, 225 };

    // ---- workspace bump allocator (fp32 units) ----
    float* Wp = (float*)d_ws;
    size_t off = 0;
    auto alloc = [&](size_t nf) { float* p = Wp + off; off += nf; return p; };

    float* xyz1     = alloc((size_t)BB * 2048 * 3);
    float* xyz2     = alloc((size_t)BB * 512 * 3);
    float* xyz3     = alloc((size_t)BB * 128 * 3);
    float* feats1   = alloc((size_t)BB * 2048 * 96);
    float* feats2   = alloc((size_t)BB * 512 * 256);
    float* feats3   = alloc((size_t)BB * 128 * 1024);
    float* fpfeats2 = alloc((size_t)BB * 512 * 256);
    float* fpfeats1 = alloc((size_t)BB * 2048 * 128);
    float* fpfeats0 = alloc((size_t)BB * NPTS * 128);
    int*   fidx     = (int*)alloc((size_t)BB * 2048);
    int*   ballidx  = (int*)alloc((size_t)BB * 2048 * 32);
    float* w3       = alloc((size_t)BB * NPTS * 3);
    int*   i3       = (int*)alloc((size_t)BB * NPTS * 3);
    float* importance = alloc((size_t)BB * NPTS);
    int*   inds     = (int*)alloc((size_t)BB * TOPKK);
    float* h_sub    = alloc((size_t)BB * TOPKK * 128);
    float* bufA     = alloc((size_t)8 * 1024 * 1024);   // 8,388,608 floats
    float* bufB     = alloc((size_t)16 * 1024 * 1024);  // 16,777,216 floats

    // ======================= SA level 1 (8192 -> 2048) =======================
    fps_kernel<<<BB, 256, 0, stream>>>(pc, fidx, NPTS, 2048);
    gather_xyz_kernel<<<cdiv(BB * 2048, 256), 256, 0, stream>>>(pc, fidx, xyz1, NPTS, 2048);
    sa_scale(pc, nullptr, 0, xyz1, NPTS, 2048, 0.01f, 16, sa0s0, 3, ballidx, bufA, bufB,
             feats1, 96, 0, stream);
    sa_scale(pc, nullptr, 0, xyz1, NPTS, 2048, 0.03f, 32, sa0s1, 3, ballidx, bufA, bufB,
             feats1, 96, 32, stream);

    // ======================= SA level 2 (2048 -> 512) ========================
    fps_kernel<<<BB, 256, 0, stream>>>(xyz1, fidx, 2048, 512);
    gather_xyz_kernel<<<cdiv(BB * 512, 256), 256, 0, stream>>>(xyz1, fidx, xyz2, 2048, 512);
    sa_scale(xyz1, feats1, 96, xyz2, 2048, 512, 0.025f, 16, sa1s0, 3, ballidx, bufA, bufB,
             feats2, 256, 0, stream);
    sa_scale(xyz1, feats1, 96, xyz2, 2048, 512, 0.05f, 32, sa1s1, 3, ballidx, bufA, bufB,
             feats2, 256, 128, stream);

    // ======================= SA level 3 (512 -> 128) =========================
    fps_kernel<<<BB, 256, 0, stream>>>(xyz2, fidx, 512, 128);
    gather_xyz_kernel<<<cdiv(BB * 128, 256), 256, 0, stream>>>(xyz2, fidx, xyz3, 512, 128);
    sa_scale(xyz2, feats2, 256, xyz3, 512, 128, 0.1f, 16, sa2s0, 3, ballidx, bufA, bufB,
             feats3, 1024, 0, stream);
    sa_scale(xyz2, feats2, 256, xyz3, 512, 128, 0.15f, 32, sa2s1, 3, ballidx, bufA, bufB,
             feats3, 1024, 512, stream);

    // ======================= FP 0: (512 <- 128) ==============================
    three_nn_kernel<<<cdiv(BB * 512, 256), 256, 0, stream>>>(xyz2, xyz3, 512, 128, w3, i3);
    interp_concat_kernel<<<cdiv(BB * 512 * 1280, 256), 256, 0, stream>>>(
        feats3, w3, i3, feats2, bufA, 512, 128, 1024, 256);
    run_chain(bufA, bufA, bufB, fpfeats2, BB * 512, fp0, 2, /*relu*/1, stream);

    // ======================= FP 1: (2048 <- 512) =============================
    three_nn_kernel<<<cdiv(BB * 2048, 256), 256, 0, stream>>>(xyz1, xyz2, 2048, 512, w3, i3);
    interp_concat_kernel<<<cdiv(BB * 2048 * 352, 256), 256, 0, stream>>>(
        fpfeats2, w3, i3, feats1, bufA, 2048, 512, 256, 96);
    run_chain(bufA, bufA, bufB, fpfeats1, BB * 2048, fp1, 2, /*relu*/1, stream);

    // ======================= FP 2: (8192 <- 2048) ============================
    three_nn_kernel<<<cdiv(BB * NPTS, 256), 256, 0, stream>>>(pc, xyz1, NPTS, 2048, w3, i3);
    interp_concat_kernel<<<cdiv(BB * NPTS * 128, 256), 256, 0, stream>>>(
        fpfeats1, w3, i3, nullptr, bufA, NPTS, 2048, 128, 0);
    run_chain(bufA, bufA, bufB, fpfeats0, BB * NPTS, fp2, 2, /*relu*/1, stream);

    // =================== importance head + top-k + FC =======================
    float* f2 = run_chain(fpfeats0, bufA, bufB, nullptr, BB * NPTS, imp, 2,
                          /*leaky 0.1*/2, stream);
    att_kernel<<<cdiv(BB * NPTS, 256), 256, 0, stream>>>(f2, att_w, att_b, importance,
                                                         BB * NPTS, 64);
    topk_kernel<<<BB, 256, 0, stream>>>(importance, inds, NPTS, TOPKK);
    gather_rows_kernel<<<cdiv(BB * TOPKK * 128, 256), 256, 0, stream>>>(
        fpfeats0, inds, h_sub, NPTS, TOPKK, 128);

    float* out = (float*)d_out;
    launch_gemm(h_sub, fc, out, BB * TOPKK, /*none*/0, stream);       // x: 460800
    i2f_kernel<<<cdiv(BB * TOPKK, 256), 256, 0, stream>>>(inds, out + 460800,
                                                          BB * TOPKK); // inds: 2048
    copyf_kernel<<<cdiv(BB * NPTS, 256), 256, 0, stream>>>(importance, out + 462848,
                                                           BB * NPTS); // importance
}